// Motion_Temporal_Excitation_4913442587332
// MI455X (gfx1250) — compile-verified
//
#include <hip/hip_runtime.h>

// CDNA5 / gfx1250. Fused Motion-Temporal-Excitation, fp32 end-to-end.
// - squeeze GEMM + conv1 GEMM on V_WMMA_F32_16X16X4_F32 (full fp32 precision;
//   kernel is HBM-bound at ~315 MB total traffic, so compute precision is free)
// - x tile staged with GLOBAL_LOAD_ASYNC_TO_LDS_B32 (ASYNCcnt path, no VGPR
//   round-trip), non-temporal stores for the write-once output.

typedef float v2f __attribute__((ext_vector_type(2)));
typedef float v8f __attribute__((ext_vector_type(8)));

#define N_    128
#define C_    256
#define T_    48
#define V_    25
#define CR_   16
#define SEG_  3
#define PIX   (SEG_ * V_)     // 75 pixels per (n, segment)
#define PIXP  80              // padded row (bank-friendly, room for 5 zero cols)
#define SEGS  (T_ / SEG_)     // 16 segments per sample
#define TVC   (T_ * V_)       // 1200

// LDS layout (in floats)
#define OFF_XS    0                        // x tile       [256][80]  = 20480
#define OFF_WSQ   (OFF_XS  + C_  * PIXP)   // w_sq         [16][256]  =  4096
#define OFF_YS    (OFF_WSQ + CR_ * C_)     // y (post-BN)  [16][80]   =  1280
#define OFF_Y1S   (OFF_YS  + CR_ * PIXP)   // y1           [16][80]   =  1280
#define OFF_WC1   (OFF_Y1S + CR_ * PIXP)   // w_c1         [16][16]   =   256
#define OFF_SA    (OFF_WC1 + CR_ * CR_)    // BN scale     [16]
#define OFF_TA    (OFF_SA  + CR_)          // BN shift(+b) [16]
#define OFF_BC1   (OFF_TA  + CR_)          // b_c1         [16]
#define OFF_M     (OFF_BC1 + CR_)          // joint-mean   [16][3]
#define OFF_GATE  (OFF_M   + CR_ * SEG_)   // gate         [256][3]
#define SMEM_FLOATS (OFF_GATE + C_ * SEG_) // 28256 floats = 113 KB -> 2 WGs/WGP

extern "C" __global__ __launch_bounds__(256)
void mte_fused_kernel(const float* __restrict__ x,
                      const float* __restrict__ w_sq,  const float* __restrict__ b_sq,
                      const float* __restrict__ gamma, const float* __restrict__ beta,
                      const float* __restrict__ rmean, const float* __restrict__ rvar,
                      const float* __restrict__ w_c1,  const float* __restrict__ b_c1,
                      const float* __restrict__ w_ex,  const float* __restrict__ b_ex,
                      float* __restrict__ out)
{
    extern __shared__ float sm[];
    const int tid  = threadIdx.x;
    const int n    = blockIdx.x / SEGS;
    const int s    = blockIdx.x % SEGS;
    const long base = (long)n * (C_ * TVC) + (long)s * PIX;  // x/out offset of this tile

    // ---------------- Phase 0a: async-stage x tile into LDS -------------------
    // GLOBAL_LOAD_ASYNC_TO_LDS_B32: per-lane global addr -> per-lane LDS addr,
    // tracked by ASYNCcnt; bypasses VGPRs entirely.
    for (int idx = tid; idx < C_ * PIX; idx += 256) {
        const int c = idx / PIX;
        const int p = idx - c * PIX;
        const float* gp = x + base + (long)c * TVC + p;
        // LDS byte address = low 32 bits of the generic shared pointer
        const unsigned lp =
            (unsigned)(unsigned long long)(sm + OFF_XS + c * PIXP + p);
        asm volatile("global_load_async_to_lds_b32 %0, %1, off"
                     :: "v"(lp), "v"(gp) : "memory");
    }
    {   // zero the 5 pad columns so the tail WMMA tile is clean (plain DS)
        const int c = tid;          // 256 threads <-> 256 channels
        #pragma unroll
        for (int p = PIX; p < PIXP; ++p) sm[OFF_XS + c * PIXP + p] = 0.0f;
    }
    // ---------------- Phase 0b: weights + folded BN affine --------------------
    for (int idx = tid; idx < CR_ * C_; idx += 256) sm[OFF_WSQ + idx] = w_sq[idx];
    sm[OFF_WC1 + tid] = w_c1[tid];  // exactly 256 elements
    if (tid < CR_) {
        const float sc = gamma[tid] * rsqrtf(rvar[tid] + 1e-5f);
        sm[OFF_SA  + tid] = sc;
        sm[OFF_TA  + tid] = beta[tid] + (b_sq[tid] - rmean[tid]) * sc; // bias folded
        sm[OFF_BC1 + tid] = b_c1[tid];
    }
    // each wave drains its own async copies, then the barrier publishes LDS
    asm volatile("s_wait_asynccnt 0x0" ::: "memory");
    __syncthreads();

    const int wave = tid >> 5;
    const int lane = tid & 31;
    const int half = lane >> 4;   // 0: lanes 0-15, 1: lanes 16-31
    const int l15  = lane & 15;   // M row (for A) / N col (for B)
    const int k2   = 2 * half;    // K sub-offset per ISA 16x16x4 f32 layout

    // ---------------- Phase 1: squeeze GEMM y[16 x 75] = w_sq (16x256) @ X ---
    // 5 waves, one 16-pixel tile each; 64 x v_wmma_f32_16x16x4_f32 per wave.
    // (wave-uniform branch => EXEC all-ones inside, as WMMA requires)
    if (wave < 5) {
        const int pixb = wave * 16;
        v8f acc = {};
        #pragma unroll 8
        for (int k = 0; k < C_; k += 4) {
            v2f a, b;
            a.x = sm[OFF_WSQ + l15 * C_ + k + k2];         // A: VGPR0 -> K = k   (+2 hi half)
            a.y = sm[OFF_WSQ + l15 * C_ + k + 1 + k2];     //    VGPR1 -> K = k+1 (+2 hi half)
            b.x = sm[OFF_XS + (k + k2)     * PIXP + pixb + l15];
            b.y = sm[OFF_XS + (k + 1 + k2) * PIXP + pixb + l15];
            acc = __builtin_amdgcn_wmma_f32_16x16x4_f32(
                      false, a, false, b, (short)0, acc, false, false);
        }
        #pragma unroll
        for (int j = 0; j < 8; ++j) {                      // C/D: VGPR j -> row j, j+8
            const int r = j + 8 * half;
            sm[OFF_YS + r * PIXP + pixb + l15] =
                acc[j] * sm[OFF_SA + r] + sm[OFF_TA + r];  // conv bias + BN affine
        }
    }
    __syncthreads();

    // ---------------- Phase 2: conv1 GEMM y1 = w_c1 (16x16) @ y ---------------
    if (wave < 5) {
        const int pixb = wave * 16;
        v8f acc = {};
        #pragma unroll
        for (int k = 0; k < CR_; k += 4) {
            v2f a, b;
            a.x = sm[OFF_WC1 + l15 * CR_ + k + k2];
            a.y = sm[OFF_WC1 + l15 * CR_ + k + 1 + k2];
            b.x = sm[OFF_YS + (k + k2)     * PIXP + pixb + l15];
            b.y = sm[OFF_YS + (k + 1 + k2) * PIXP + pixb + l15];
            acc = __builtin_amdgcn_wmma_f32_16x16x4_f32(
                      false, a, false, b, (short)0, acc, false, false);
        }
        #pragma unroll
        for (int j = 0; j < 8; ++j) {
            const int r = j + 8 * half;
            sm[OFF_Y1S + r * PIXP + pixb + l15] = acc[j] + sm[OFF_BC1 + r];
        }
    }
    __syncthreads();

    // ---------------- Phase 3: temporal diff + mean over joints ---------------
    // diff[f] = y1[f+1] - y[f] (f = 0,1), zero for f = 2; m = mean over V.
    if (tid < CR_ * SEG_) {
        const int r = tid / SEG_;
        const int f = tid - r * SEG_;
        float sum = 0.0f;
        if (f < SEG_ - 1) {
            #pragma unroll
            for (int v = 0; v < V_; ++v)
                sum += sm[OFF_Y1S + r * PIXP + (f + 1) * V_ + v]
                     - sm[OFF_YS  + r * PIXP +  f      * V_ + v];
        }
        sm[OFF_M + tid] = sum * (1.0f / V_);
    }
    __syncthreads();

    // ---------------- Phase 4: expand 16->256 + sigmoid gate ------------------
    {
        const int c = tid;
        float g0 = b_ex[c], g1 = g0, g2 = g0;
        #pragma unroll
        for (int r = 0; r < CR_; ++r) {
            const float w = w_ex[c * CR_ + r];
            g0 += w * sm[OFF_M + r * SEG_ + 0];
            g1 += w * sm[OFF_M + r * SEG_ + 1];
            g2 += w * sm[OFF_M + r * SEG_ + 2];
        }
        sm[OFF_GATE + c * SEG_ + 0] = 1.0f / (1.0f + __expf(-g0));
        sm[OFF_GATE + c * SEG_ + 1] = 1.0f / (1.0f + __expf(-g1));
        sm[OFF_GATE + c * SEG_ + 2] = 1.0f / (1.0f + __expf(-g2));
    }
    __syncthreads();

    // ---------------- Phase 5: out = x * gate (coalesced, non-temporal) -------
    for (int idx = tid; idx < C_ * PIX; idx += 256) {
        const int c = idx / PIX;
        const int p = idx - c * PIX;
        const int f = p / V_;
        const float v = sm[OFF_XS + c * PIXP + p] * sm[OFF_GATE + c * SEG_ + f];
        __builtin_nontemporal_store(v, &out[base + (long)c * TVC + p]);
    }
}

extern "C" void kernel_launch(void* const* d_in, const int* in_sizes, int n_in,
                              void* d_out, int out_size, void* d_ws, size_t ws_size,
                              hipStream_t stream) {
    const float* x     = (const float*)d_in[0];
    const float* w_sq  = (const float*)d_in[1];
    const float* b_sq  = (const float*)d_in[2];
    const float* gamma = (const float*)d_in[3];
    const float* beta  = (const float*)d_in[4];
    const float* rmean = (const float*)d_in[5];
    const float* rvar  = (const float*)d_in[6];
    const float* w_c1  = (const float*)d_in[7];
    const float* b_c1  = (const float*)d_in[8];
    const float* w_ex  = (const float*)d_in[9];
    const float* b_ex  = (const float*)d_in[10];
    float* out = (float*)d_out;

    const dim3 grid(N_ * SEGS);          // 2048 workgroups = one per (n, segment)
    const dim3 block(256);               // 8 waves (wave32)
    const size_t shmem = SMEM_FLOATS * sizeof(float);   // ~113 KB -> 2 WGs/WGP

    mte_fused_kernel<<<grid, block, shmem, stream>>>(
        x, w_sq, b_sq, gamma, beta, rmean, rvar, w_c1, b_c1, w_ex, b_ex, out);
}